// DGCNN_15779709845546
// MI455X (gfx1250) — compile-verified
//
#include <hip/hip_runtime.h>

typedef float v2f __attribute__((ext_vector_type(2)));
typedef float v8f __attribute__((ext_vector_type(8)));

#define NN 100000
#define NE 1600000
#define NG 1000
#define DF 32
#define CATF 129
#define BN_EPS 1e-5f

// ---------------- utility ----------------
__global__ void zero_f32(float* p, int n) {
  int i = blockIdx.x * blockDim.x + threadIdx.x;
  if (i < n) p[i] = 0.0f;
}

__global__ void deg_kernel(const int* __restrict__ dst, float* __restrict__ deg) {
  int e = blockIdx.x * blockDim.x + threadIdx.x;
  if (e < NE) atomicAdd(&deg[dst[e]], 1.0f);
}

__global__ void cnt_kernel(const int* __restrict__ gid, float* __restrict__ cnt) {
  int n = blockIdx.x * blockDim.x + threadIdx.x;
  if (n < NN) atomicAdd(&cnt[gid[n]], 1.0f);
}

// scatter-add x[src] into agg[dst]; one thread per (edge, 4-channel chunk)
__global__ void scatter_kernel(const float* __restrict__ x, const int* __restrict__ src,
                               const int* __restrict__ dst, float* __restrict__ agg) {
  int i = blockIdx.x * blockDim.x + threadIdx.x;
  if (i >= NE * 8) return;
  int e = i >> 3;
  int q = (i & 7) * 4;
  int s = src[e];
  int d = dst[e];
  float4 v = *(const float4*)&x[s * DF + q];
  float* o = &agg[d * DF + q];
  atomicAdd(o + 0, v.x);
  atomicAdd(o + 1, v.y);
  atomicAdd(o + 2, v.z);
  atomicAdd(o + 3, v.w);
}

__global__ void meandiv_kernel(float* __restrict__ agg, const float* __restrict__ deg) {
  int i = blockIdx.x * blockDim.x + threadIdx.x;
  if (i < NN * DF) agg[i] = agg[i] / fmaxf(deg[i / DF], 1.0f);
}

// ---------------- WMMA GEMM: Y[N,32] = A[N,32] @ W[32,32] + bias ----------------
// One wave handles a 16-row tile. K=32 -> 8 chained V_WMMA_F32_16X16X4_F32 per
// 16-col output tile; two accumulators cover the 32 output columns.
// A-layout (16x4 f32): lanes 0-15 M=lane, K={k0,k0+1}; lanes 16-31 M=lane-16, K={k0+2,k0+3}
// B-layout (4x16 f32): mirrored, N across lanes.
// D-layout (16x16 f32): VGPR r: lanes 0-15 -> M=r, lanes 16-31 -> M=8+r; N=lane&15.
__launch_bounds__(256)
__global__ void gemm32_wmma(const float* __restrict__ A, const float* __restrict__ W,
                            const float* __restrict__ bias, float* __restrict__ Y) {
  int wave = (blockIdx.x * blockDim.x + threadIdx.x) >> 5;
  int lane = threadIdx.x & 31;
  const int ntiles = NN / 16;  // 6250, exact
  if (wave >= ntiles) return;  // wave-uniform: EXEC stays all-1s for WMMA
  int hf  = lane >> 4;
  int lid = lane & 15;
  int row0 = wave * 16;

  float b0 = bias[lid];
  float b1 = bias[16 + lid];
  v8f c0, c1;
#pragma unroll
  for (int r = 0; r < 8; ++r) { c0[r] = b0; c1[r] = b1; }

#pragma unroll
  for (int kt = 0; kt < 8; ++kt) {
    int k = kt * 4 + 2 * hf;
    v2f a = *(const v2f*)&A[(row0 + lid) * DF + k];
    v2f bA, bB;
    bA.x = W[k * DF + lid];
    bA.y = W[(k + 1) * DF + lid];
    bB.x = W[k * DF + 16 + lid];
    bB.y = W[(k + 1) * DF + 16 + lid];
    c0 = __builtin_amdgcn_wmma_f32_16x16x4_f32(false, a, false, bA, (short)0, c0, false, false);
    c1 = __builtin_amdgcn_wmma_f32_16x16x4_f32(false, a, false, bB, (short)0, c1, false, false);
  }

#pragma unroll
  for (int r = 0; r < 8; ++r) {
    int row = row0 + 8 * hf + r;
    Y[row * DF + lid]      = c0[r];
    Y[row * DF + 16 + lid] = c1[r];
  }
}

// ---------------- BatchNorm statistics (sum, sumsq per channel) ----------------
// requires blockDim.x % C == 0 (C in {1,32}); stats[0..C-1]=sum, stats[32..32+C-1]=sumsq
__global__ void bnstats_kernel(const float* __restrict__ Y, int C, float* __restrict__ stats) {
  __shared__ float ssum[DF];
  __shared__ float ssq[DF];
  int t = threadIdx.x;
  if (t < C) { ssum[t] = 0.0f; ssq[t] = 0.0f; }
  __syncthreads();
  int c = t % C;
  float s = 0.0f, q = 0.0f;
  int total = NN * C;
  for (int i = blockIdx.x * blockDim.x + t; i < total; i += gridDim.x * blockDim.x) {
    float v = Y[i];
    s += v;
    q += v * v;
  }
  atomicAdd(&ssum[c], s);   // ds_add_f32
  atomicAdd(&ssq[c], q);
  __syncthreads();
  if (t < C) {
    atomicAdd(&stats[t], ssum[t]);
    atomicAdd(&stats[DF + t], ssq[t]);
  }
}

__global__ void bnfinal_kernel(const float* __restrict__ stats, const float* __restrict__ g,
                               const float* __restrict__ b, int C, float* __restrict__ ss) {
  int c = threadIdx.x;
  if (c < C) {
    float mean = stats[c] * (1.0f / NN);
    float var  = stats[DF + c] * (1.0f / NN) - mean * mean;
    float rs   = rsqrtf(var + BN_EPS);
    float sc   = g[c] * rs;
    ss[c]      = sc;
    ss[DF + c] = b[c] - mean * sc;
  }
}

// BN apply + ReLU, write next-layer activations, and scatter-add into pooled hg
__global__ void bnapply_pool(const float* __restrict__ Y, const float* __restrict__ ss,
                             const int* __restrict__ gid, int C, int catoff,
                             float* __restrict__ Xout, float* __restrict__ hg) {
  int i = blockIdx.x * blockDim.x + threadIdx.x;
  if (i >= NN * C) return;
  int n = i / C;
  int c = i % C;
  float v = fmaxf(Y[i] * ss[c] + ss[DF + c], 0.0f);
  if (Xout) Xout[i] = v;
  atomicAdd(&hg[gid[n] * CATF + catoff + c], v);
}

// raw pooling of input features h into hg[:, 0:32]
__global__ void pool_raw(const float* __restrict__ X, const int* __restrict__ gid,
                         float* __restrict__ hg) {
  int i = blockIdx.x * blockDim.x + threadIdx.x;
  if (i >= NN * DF) return;
  int n = i / DF;
  int c = i % DF;
  atomicAdd(&hg[gid[n] * CATF + c], X[i]);
}

// layer-3 conv: y[n] = dot32(agg[n], w) + b (output width 1)
__global__ void dot32_kernel(const float* __restrict__ A, const float* __restrict__ w,
                             const float* __restrict__ b, float* __restrict__ y) {
  int n = blockIdx.x * blockDim.x + threadIdx.x;
  if (n >= NN) return;
  float s = b[0];
#pragma unroll
  for (int k = 0; k < DF; k += 4) {
    float4 a  = *(const float4*)&A[n * DF + k];
    float4 wv = *(const float4*)&w[k];
    s += a.x * wv.x + a.y * wv.y + a.z * wv.z + a.w * wv.w;
  }
  y[n] = s;
}

__global__ void pooldiv_kernel(float* __restrict__ hg, const float* __restrict__ cnt) {
  int i = blockIdx.x * blockDim.x + threadIdx.x;
  if (i >= NG * CATF) return;
  hg[i] = hg[i] / fmaxf(cnt[i / CATF], 1.0f);
}

// small dense layer: Y[g,j] = act(sum_k X[g,k]*W[k,j] + b[j])
__global__ void dense_kernel(const float* __restrict__ X, const float* __restrict__ W,
                             const float* __restrict__ b, float* __restrict__ Y,
                             int Gn, int K, int J, int relu) {
  int i = blockIdx.x * blockDim.x + threadIdx.x;
  if (i >= Gn * J) return;
  int g = i / J;
  int j = i % J;
  float s = b[j];
  for (int k = 0; k < K; ++k) s += X[g * K + k] * W[k * J + j];
  if (relu) s = fmaxf(s, 0.0f);
  Y[i] = s;
}

// ---------------- launcher ----------------
extern "C" void kernel_launch(void* const* d_in, const int* in_sizes, int n_in,
                              void* d_out, int out_size, void* d_ws, size_t ws_size,
                              hipStream_t stream) {
  const float* h   = (const float*)d_in[0];
  const int*   src = (const int*)d_in[1];
  const int*   dst = (const int*)d_in[2];
  const int*   gid = (const int*)d_in[3];
  const float* conv_w[4] = {(const float*)d_in[4],  (const float*)d_in[8],
                            (const float*)d_in[12], (const float*)d_in[16]};
  const float* conv_b[4] = {(const float*)d_in[5],  (const float*)d_in[9],
                            (const float*)d_in[13], (const float*)d_in[17]};
  const float* bn_g[4]   = {(const float*)d_in[6],  (const float*)d_in[10],
                            (const float*)d_in[14], (const float*)d_in[18]};
  const float* bn_b[4]   = {(const float*)d_in[7],  (const float*)d_in[11],
                            (const float*)d_in[15], (const float*)d_in[19]};
  const float* mlp_w0 = (const float*)d_in[20];
  const float* mlp_b0 = (const float*)d_in[21];
  const float* mlp_w1 = (const float*)d_in[22];
  const float* mlp_b1 = (const float*)d_in[23];
  const float* mlp_w2 = (const float*)d_in[24];
  const float* mlp_b2 = (const float*)d_in[25];

  float* ws = (float*)d_ws;
  size_t o = 0;
  float* deg   = ws + o; o += NN;
  float* cnt   = ws + o; o += NG;
  float* agg   = ws + o; o += (size_t)NN * DF;
  float* ybuf  = ws + o; o += (size_t)NN * DF;
  float* xA    = ws + o; o += (size_t)NN * DF;
  float* xB    = ws + o; o += (size_t)NN * DF;
  float* stats = ws + o; o += 2 * DF;
  float* ss    = ws + o; o += 2 * DF;
  float* hg    = ws + o; o += (size_t)NG * CATF;
  float* h1    = ws + o; o += (size_t)NG * 128;
  float* h2    = ws + o; o += (size_t)NG * 64;

  const int B = 256;
  auto blk = [](int n, int b) { return (n + b - 1) / b; };

  // init: degrees, per-graph counts, pooled buffer, pool raw h
  zero_f32<<<blk(NN, B), B, 0, stream>>>(deg, NN);
  zero_f32<<<blk(NG, B), B, 0, stream>>>(cnt, NG);
  zero_f32<<<blk(NG * CATF, B), B, 0, stream>>>(hg, NG * CATF);
  deg_kernel<<<blk(NE, B), B, 0, stream>>>(dst, deg);
  cnt_kernel<<<blk(NN, B), B, 0, stream>>>(gid, cnt);
  pool_raw<<<blk(NN * DF, B), B, 0, stream>>>(h, gid, hg);

  // layers 0..2 (32 -> 32), WMMA GEMM path
  const float* xin = h;
  float* xping[2] = {xA, xB};
  for (int l = 0; l < 3; ++l) {
    zero_f32<<<blk(NN * DF, B), B, 0, stream>>>(agg, NN * DF);
    scatter_kernel<<<blk(NE * 8, B), B, 0, stream>>>(xin, src, dst, agg);
    meandiv_kernel<<<blk(NN * DF, B), B, 0, stream>>>(agg, deg);
    gemm32_wmma<<<blk(NN / 16, 8), B, 0, stream>>>(agg, conv_w[l], conv_b[l], ybuf);
    zero_f32<<<1, 64, 0, stream>>>(stats, 2 * DF);
    bnstats_kernel<<<512, B, 0, stream>>>(ybuf, DF, stats);
    bnfinal_kernel<<<1, DF, 0, stream>>>(stats, bn_g[l], bn_b[l], DF, ss);
    bnapply_pool<<<blk(NN * DF, B), B, 0, stream>>>(ybuf, ss, gid, DF, DF * (l + 1),
                                                    xping[l & 1], hg);
    xin = xping[l & 1];
  }

  // layer 3 (32 -> 1)
  zero_f32<<<blk(NN * DF, B), B, 0, stream>>>(agg, NN * DF);
  scatter_kernel<<<blk(NE * 8, B), B, 0, stream>>>(xin, src, dst, agg);
  meandiv_kernel<<<blk(NN * DF, B), B, 0, stream>>>(agg, deg);
  dot32_kernel<<<blk(NN, B), B, 0, stream>>>(agg, conv_w[3], conv_b[3], ybuf);
  zero_f32<<<1, 64, 0, stream>>>(stats, 2 * DF);
  bnstats_kernel<<<512, B, 0, stream>>>(ybuf, 1, stats);
  bnfinal_kernel<<<1, DF, 0, stream>>>(stats, bn_g[3], bn_b[3], 1, ss);
  bnapply_pool<<<blk(NN, B), B, 0, stream>>>(ybuf, ss, gid, 1, 128, nullptr, hg);

  // per-graph mean + MLP head
  pooldiv_kernel<<<blk(NG * CATF, B), B, 0, stream>>>(hg, cnt);
  dense_kernel<<<blk(NG * 128, B), B, 0, stream>>>(hg, mlp_w0, mlp_b0, h1, NG, CATF, 128, 1);
  dense_kernel<<<blk(NG * 64, B), B, 0, stream>>>(h1, mlp_w1, mlp_b1, h2, NG, 128, 64, 1);
  dense_kernel<<<blk(NG, B), B, 0, stream>>>(h2, mlp_w2, mlp_b2, (float*)d_out, NG, 64, 1, 0);
}